// LogitsProcessorWithPacked_28973849379121
// MI455X (gfx1250) — compile-verified
//
#include <hip/hip_runtime.h>

// ---------------------------------------------------------------------------
// Per-token LoRA/delta LM-head:  out[t, v] = W[idx[t], v, :] . hs[t, :]
//   hs : [64, 2048] f32     W : [4, 32000, 2048] f32     idx : [64] i32
//   out: [64, 32000] f32
//
// MI455X reasoning: weights are ~1.05 GB fp32; at 23.3 TB/s the floor is
// ~45 us if each delta is streamed ONCE. AI = 8 FLOP/B -> memory bound. We
// group tokens by delta into <=7 padded 16-token tiles and run one
// V_WMMA_F32_16X16X4_F32 GEMM per tile (full fp32 precision).
//
// Data movement uses the CDNA5 async path: GLOBAL_LOAD_ASYNC_TO_LDS_B128
// DMAs weights/hidden HBM->LDS directly (no VGPR staging), tracked with
// ASYNCcnt; double-buffered so chunk c+1's HBM traffic overlaps chunk c's
// WMMA loop. Fragment reads use a 132-float padded stride so the 32-lane
// b64 reads hit all 64 LDS banks conflict-free (compiler fuses them into
// ds_load_2addr_b64 pairs).
// ---------------------------------------------------------------------------

#define NUM_TOKENS 64
#define HIDDEN     2048
#define VOCAB      32000
#define NDELTA     4

#define KC       128           // K-chunk per LDS stage
#define STR      132           // padded LDS row stride in floats (== 4 mod 64)
#define WAVES    8             // waves per block (wave32 -> 256 threads)
#define BLOCKSZ  (WAVES * 32)
#define VROWS    (WAVES * 16)  // vocab rows per block = 128
#define NCHUNK   (HIDDEN / KC) // 16
#define MAXTILES 8             // worst-case token tiles: 61,1,1,1 -> 7

typedef __attribute__((ext_vector_type(2))) float v2f;
typedef __attribute__((ext_vector_type(8))) float v8f;

// 32-bit LDS byte offset of a generic pointer known to live in shared memory.
typedef __attribute__((address_space(3))) const void* lds_cptr_t;
static __device__ __forceinline__ unsigned lds_off_of(const void* p) {
  return (unsigned)(unsigned long long)(lds_cptr_t)p;
}

// One per-lane 16-byte HBM -> LDS async DMA (VDST = LDS byte address VGPR,
// VADDR = 64-bit global address VGPR pair). Tracked with ASYNCcnt.
static __device__ __forceinline__ void async_b128(unsigned lds_addr,
                                                  const float* gptr) {
  asm volatile("global_load_async_to_lds_b128 %0, %1, off"
               :
               : "v"(lds_addr), "v"(gptr)
               : "memory");
}

static __device__ __forceinline__ void wait_async0() {
  asm volatile("s_wait_asynccnt 0x0" ::: "memory");
}

// ---------------------------------------------------------------------------
// Kernel 1: bucket 64 tokens by delta index into padded 16-token tiles.
// ---------------------------------------------------------------------------
__global__ void build_tiles_kernel(const int* __restrict__ idx,
                                   int* __restrict__ tile_delta,
                                   int* __restrict__ tile_tok) {
  if (threadIdx.x != 0) return;
  int grouped[NUM_TOKENS];
  int pos = 0, tile = 0;
  for (int d = 0; d < NDELTA; ++d) {
    int start = pos;
    for (int t = 0; t < NUM_TOKENS; ++t)
      if (idx[t] == d) grouped[pos++] = t;
    int n = pos - start;
    for (int off = 0; off < n; off += 16) {
      tile_delta[tile] = d;
      for (int j = 0; j < 16; ++j)
        tile_tok[tile * 16 + j] = (off + j < n) ? grouped[start + off + j] : -1;
      ++tile;
    }
  }
  for (; tile < MAXTILES; ++tile) {
    tile_delta[tile] = -1;
    for (int j = 0; j < 16; ++j) tile_tok[tile * 16 + j] = -1;
  }
}

// ---------------------------------------------------------------------------
// Kernel 2: one 16-token tile x 128 vocab rows per block, K pipelined via
// async HBM->LDS DMA.
//   A (16x4, M=tokens)  from LDS hidden tile
//   B (4x16, N=vocab)   from LDS weight tile (W row-major == W^T fragment)
//   D (16x16)           token-major rows -> coalesced stores to out[t, v]
// ---------------------------------------------------------------------------
__global__ void lmhead_swap_wmma_kernel(const float* __restrict__ hs,
                                        const float* __restrict__ w,
                                        const int* __restrict__ tile_delta,
                                        const int* __restrict__ tile_tok,
                                        float* __restrict__ out) {
  extern __shared__ float smem[];
  float* sh_h = smem;                  // [2][16][STR] hidden double buffer
  float* sh_w = smem + 2 * 16 * STR;   // [2][WAVES][16][STR] weight buffers
  int* sh_tok = (int*)(smem + 2 * 16 * STR + 2 * WAVES * 16 * STR);

  const int slot = blockIdx.y;
  const int d = tile_delta[slot];
  if (d < 0) return;  // uniform across block: safe early-exit

  const int tid  = threadIdx.x;
  const int wave = tid >> 5;
  const int lane = tid & 31;
  const int hi   = lane >> 4;   // upper half-wave handles K+2/K+3
  const int ln   = lane & 15;

  if (tid < 16) sh_tok[tid] = tile_tok[slot * 16 + tid];

  // Weight base for this wave's 16 vocab rows of delta d.
  const size_t wbase =
      ((size_t)d * VOCAB + (size_t)blockIdx.x * VROWS + (size_t)wave * 16) *
      (size_t)HIDDEN;

  // Hidden loader mapping: 16 threads per token row, 8 contiguous floats each.
  const int th = tid >> 4;           // token row 0..15
  const int tk = (tid & 15) * 8;     // k offset within chunk
  const int tokh = tile_tok[slot * 16 + th];
  // Padded rows alias token 0: garbage lands only in D rows never stored.
  const int tok_src = (tokh >= 0) ? tokh : 0;
  const float* hrow = hs + (size_t)tok_src * HIDDEN + tk;

  // ---- prologue: fire chunk 0 async DMAs into buffer 0 ----
  {
    float* hbuf = sh_h;
    float* wbuf = sh_w + (size_t)wave * 16 * STR;
#pragma unroll
    for (int r = 0; r < 16; ++r)
      async_b128(lds_off_of(wbuf + r * STR + lane * 4),
                 w + wbase + (size_t)r * HIDDEN + lane * 4);
    async_b128(lds_off_of(hbuf + th * STR + tk), hrow);
    async_b128(lds_off_of(hbuf + th * STR + tk + 4), hrow + 4);
  }

  v8f acc = {};

  for (int c = 0; c < NCHUNK; ++c) {
    const int cb = c & 1;
    float* hbuf = sh_h + cb * 16 * STR;
    float* wbuf = sh_w + (size_t)(cb * WAVES + wave) * 16 * STR;

    // Only chunk c's DMAs are outstanding here: drain them, then barrier so
    // every wave's weight strip + hidden slice for chunk c is visible.
    // Passing the barrier also proves all waves finished compute(c-1), so
    // buffer (c+1)&1 is free for the DMAs issued below.
    wait_async0();
    __syncthreads();

    if (c + 1 < NCHUNK) {
      const int k0 = (c + 1) * KC;
      const int nb = (c + 1) & 1;
      float* hnext = sh_h + nb * 16 * STR;
      float* wnext = sh_w + (size_t)(nb * WAVES + wave) * 16 * STR;
#pragma unroll
      for (int r = 0; r < 16; ++r)
        async_b128(lds_off_of(wnext + r * STR + lane * 4),
                   w + wbase + (size_t)r * HIDDEN + k0 + lane * 4);
      async_b128(lds_off_of(hnext + th * STR + tk), hrow + k0);
      async_b128(lds_off_of(hnext + th * STR + tk + 4), hrow + k0 + 4);
    }

    // Fragment base: element (row=ln, k = kk + 2*hi). STR==132 (4 mod 64)
    // makes the 32-lane b64 reads cover all 64 LDS banks exactly once.
    const float* ha = hbuf + ln * STR + 2 * hi;
    const float* wb = wbuf + ln * STR + 2 * hi;
#pragma unroll
    for (int kk = 0; kk < KC; kk += 4) {
      v2f a = *(const v2f*)(ha + kk);  // A: hidden, M=token ln
      v2f b = *(const v2f*)(wb + kk);  // B: W^T,   N=vocab ln
      acc = __builtin_amdgcn_wmma_f32_16x16x4_f32(
          /*neg_a=*/false, a, /*neg_b=*/false, b,
          /*c_mod=*/(short)0, acc, /*reuse_a=*/false, /*reuse_b=*/false);
    }
  }

  // D layout: VGPR g -> row M = g + 8*hi (token), lane -> col N (vocab).
  // Lanes 0-15 / 16-31 each store 64 contiguous bytes -> coalesced.
  const size_t vcol = (size_t)blockIdx.x * VROWS + (size_t)wave * 16 + ln;
#pragma unroll
  for (int g = 0; g < 8; ++g) {
    const int m = g + 8 * hi;
    const int tok = sh_tok[m];
    if (tok >= 0) out[(size_t)tok * VOCAB + vcol] = acc[g];
  }
}

// ---------------------------------------------------------------------------
extern "C" void kernel_launch(void* const* d_in, const int* in_sizes, int n_in,
                              void* d_out, int out_size, void* d_ws,
                              size_t ws_size, hipStream_t stream) {
  const float* hs  = (const float*)d_in[0];  // [64, 2048]
  const float* w   = (const float*)d_in[1];  // [4, 32000, 2048]
  const int*   idx = (const int*)d_in[2];    // [64]
  float* out = (float*)d_out;                // [64, 32000]

  int* tile_delta = (int*)d_ws;              // [8]
  int* tile_tok   = tile_delta + MAXTILES;   // [8][16]

  build_tiles_kernel<<<1, 32, 0, stream>>>(idx, tile_delta, tile_tok);

  const size_t smem_bytes =
      (size_t)(2 * 16 * STR + 2 * WAVES * 16 * STR) * sizeof(float) +
      16 * sizeof(int);  // ~152 KB, within the 320 KB WGP LDS budget

  dim3 grid(VOCAB / VROWS, MAXTILES);  // 250 x 8 (invalid slots exit instantly)
  lmhead_swap_wmma_kernel<<<grid, BLOCKSZ, smem_bytes, stream>>>(
      hs, w, tile_delta, tile_tok, out);
}